// QuadraticSplineStack_55980603736339
// MI455X (gfx1250) — compile-verified
//
#include <hip/hip_runtime.h>
#include <math.h>

// =====================================================================
// QuadraticSplineStack forward for MI455X (gfx1250, wave32).
//
// Pipeline (all on `stream`, deterministic output values):
//  K1 zero_counts   : counts[g] = 0
//  K2 hist          : counts[g] += 1 per element (global atomics)
//  K3 scan_starts   : exclusive scan -> starts[G+1], cursor[G]
//  K4 scatter       : ticketed counting-sort: perm[pos]=i, xs[pos]=x[i]
//  K5 build_tables  : per-gene spline tables (gene-major, 896 f32/gene)
//  K6 eval_sorted   : one block per gene; gene's tables -> LDS (3.6 KB),
//                     evaluate its ~800 elements fully out of LDS,
//                     scatter results to original positions (NT stores).
//
// ws layout: [tabs: G*896 f32][counts: G][starts: G+1][cursor: G]
//            [perm: n i32][xs: n f32]   (~50 MB for G=5000, n=4M)
// =====================================================================

#define PARAM_STRIDE 445  // 128 + 127 + 64 + 63 + 32 + 31
// per-gene table block: stage0 @0 (4*128), stage1 @512 (4*64), stage2 @768 (4*32)
#define TAB_STRIDE 896

// ---------------- block helpers (blockDim.x == 128) ----------------
__device__ __forceinline__ float blk_reduce_max(float v, float* s) {
  const int t = threadIdx.x;
  s[t] = v; __syncthreads();
#pragma unroll
  for (int off = 64; off >= 1; off >>= 1) {
    if (t < off) s[t] = fmaxf(s[t], s[t + off]);
    __syncthreads();
  }
  float r = s[0]; __syncthreads();
  return r;
}

__device__ __forceinline__ float blk_reduce_sum(float v, float* s) {
  const int t = threadIdx.x;
  s[t] = v; __syncthreads();
#pragma unroll
  for (int off = 64; off >= 1; off >>= 1) {
    if (t < off) s[t] += s[t + off];
    __syncthreads();
  }
  float r = s[0]; __syncthreads();
  return r;
}

__device__ __forceinline__ float blk_incl_scan(float v, float* s) {
  const int t = threadIdx.x;
  s[t] = v; __syncthreads();
#pragma unroll
  for (int off = 1; off < 128; off <<= 1) {
    float u = (t >= off) ? s[t - off] : 0.0f;
    __syncthreads();
    s[t] += u;
    __syncthreads();
  }
  float r = s[t]; __syncthreads();
  return r;
}

// ---------------- K5: per-gene table construction ----------------
// base points at this gene's stage block: [locs N][heights N][lcdf N][widths N]
template <int N>
__device__ __forceinline__ void build_one(const float* __restrict__ row_uh,
                                          const float* __restrict__ row_uw,
                                          float* __restrict__ base,
                                          float* sScan, float* sRed, float* sH) {
  const int t = threadIdx.x;
  float* __restrict__ locs    = base;
  float* __restrict__ heights = base + N;
  float* __restrict__ lcdf    = base + 2 * N;
  float* __restrict__ widths  = base + 3 * N;

  // softmax over unnormalized widths (N-1 entries)
  float uw = (t < N - 1) ? row_uw[t] : -INFINITY;
  float m  = blk_reduce_max(uw, sRed);
  float e  = (t < N - 1) ? __expf(uw - m) : 0.0f;
  float S  = blk_reduce_sum(e, sRed);
  float w  = e / S;  // valid for t < N-1, else 0
  if (t < N) widths[t] = w;

  // locs[0]=0, locs[i]=cumsum(w)[i-1], locs[N-1]=1
  float cw   = blk_incl_scan(w, sScan);
  float excl = cw - w;
  if (t < N) locs[t] = (t == N - 1) ? 1.0f : excl;

  // normalized pdf heights
  float eh = (t < N) ? __expf(row_uh[t]) : 0.0f;
  sH[t] = eh; __syncthreads();
  float a    = (t < N - 1) ? 0.5f * (eh + sH[t + 1]) * w : 0.0f;
  float area = blk_reduce_sum(a, sRed);
  float h    = eh / area;
  if (t < N) heights[t] = h;
  __syncthreads();
  sH[t] = h; __syncthreads();

  // lcdf[0]=0, lcdf[i]=cumsum(c)[i-1], lcdf[N-1]=1
  float c  = (t < N - 1) ? 0.5f * (h + sH[t + 1]) * w : 0.0f;
  float cc = blk_incl_scan(c, sScan);
  float ec = cc - c;
  if (t < N) lcdf[t] = (t == N - 1) ? 1.0f : ec;
}

__global__ __launch_bounds__(128) void qspline_build_tables(
    const float* __restrict__ unnorm, float* __restrict__ tabs, int G) {
  __shared__ float sScan[128];
  __shared__ float sRed[128];
  __shared__ float sH[128];
  const int g = blockIdx.x;
  if (g >= G) return;
  const float* row = unnorm + (size_t)g * PARAM_STRIDE;
  float* base = tabs + (size_t)g * TAB_STRIDE;
  build_one<128>(row + 0,   row + 128, base + 0,   sScan, sRed, sH);
  build_one<64> (row + 255, row + 319, base + 512, sScan, sRed, sH);
  build_one<32> (row + 382, row + 414, base + 768, sScan, sRed, sH);
}

// ---------------- K1/K2: histogram ----------------
__global__ __launch_bounds__(256) void qspline_zero_counts(int* __restrict__ counts, int G) {
  const int i = blockIdx.x * blockDim.x + threadIdx.x;
  if (i < G) counts[i] = 0;
}

__global__ __launch_bounds__(256) void qspline_hist(const int* __restrict__ gix,
                                                    int* __restrict__ counts, int n) {
  const int i = blockIdx.x * blockDim.x + threadIdx.x;
  if (i < n) {
    int g = __builtin_nontemporal_load(gix + i);
    atomicAdd(&counts[g], 1);
  }
}

// ---------------- K3: single-block exclusive scan over G counters ----------------
__global__ __launch_bounds__(1024) void qspline_scan_starts(
    const int* __restrict__ counts, int* __restrict__ starts,
    int* __restrict__ cursor, int G) {
  __shared__ int s[1024];
  __shared__ int carry;
  const int t = threadIdx.x;
  if (t == 0) carry = 0;
  __syncthreads();
  for (int base = 0; base < G; base += 1024) {
    int v = (base + t < G) ? counts[base + t] : 0;
    s[t] = v; __syncthreads();
#pragma unroll
    for (int off = 1; off < 1024; off <<= 1) {
      int u = (t >= off) ? s[t - off] : 0;
      __syncthreads();
      s[t] += u;
      __syncthreads();
    }
    int incl  = s[t];
    int c     = carry;          // old carry (stable: writer synced below)
    int excl  = c + incl - v;
    if (base + t < G) { starts[base + t] = excl; cursor[base + t] = excl; }
    int total = s[1023];
    __syncthreads();            // all reads of carry/s done
    if (t == 0) carry = c + total;
    __syncthreads();
  }
  if (t == 0) starts[G] = carry;  // == n
}

// ---------------- K4: ticketed counting-sort scatter ----------------
__global__ __launch_bounds__(256) void qspline_scatter(
    const float* __restrict__ x, const int* __restrict__ gix,
    int* __restrict__ cursor, int* __restrict__ perm,
    float* __restrict__ xs, int n) {
  const int i = blockIdx.x * blockDim.x + threadIdx.x;
  if (i < n) {
    int g   = __builtin_nontemporal_load(gix + i);
    float v = __builtin_nontemporal_load(x + i);
    int pos = atomicAdd(&cursor[g], 1);
    perm[pos] = i;   // RT stores: stay L2-resident until K6 consumes them
    xs[pos]   = v;
  }
}

// ---------------- K6: LDS-resident spline evaluation ----------------
// t points at one stage block in LDS: [locs N][heights N][lcdf N][widths N]
template <int N, int ITERS>
__device__ __forceinline__ void spline_lds(const float* __restrict__ t,
                                           float& x, float& lad) {
  const float* locs = t;
  const float* hts  = t + N;
  const float* lc_  = t + 2 * N;
  const float* wid  = t + 3 * N;

  // searchsorted(locs, x, side='right') - 1, clamped to [0, N-2]
  int lo = 0, hi = N;
#pragma unroll
  for (int it = 0; it < ITERS; ++it) {
    int mid = (lo + hi) >> 1;
    mid = min(max(mid, 0), N - 1);
    bool pred = locs[mid] <= x;
    lo = pred ? (mid + 1) : lo;
    hi = pred ? hi : mid;
  }
  int b = min(max(lo - 1, 0), N - 2);

  float w  = wid[b];
  float lh = hts[b];
  float rh = hts[b + 1];
  float bl = locs[b];
  float lc = lc_[b];

  float alpha = (x - bl) / w;
  float out = (0.5f * (rh - lh) * w) * alpha * alpha + (lh * w) * alpha + lc;
  out = fminf(fmaxf(out, 0.0f), 1.0f);
  lad += __logf(alpha * (rh - lh) + lh);
  x = out;
}

__global__ __launch_bounds__(256) void qspline_eval_sorted(
    const float* __restrict__ xs, const int* __restrict__ perm,
    const int* __restrict__ starts, const float* __restrict__ tabs,
    float* __restrict__ out, int n) {
  __shared__ float tab[TAB_STRIDE];
  const int g = blockIdx.x;

  // stage this gene's full table set into LDS (896 f32 = 3.6 KB)
  const float* src = tabs + (size_t)g * TAB_STRIDE;
  for (int j = threadIdx.x; j < TAB_STRIDE; j += 256) tab[j] = src[j];
  const int s = starts[g];
  const int e = starts[g + 1];
  __syncthreads();

  for (int j = s + threadIdx.x; j < e; j += 256) {
    float xv  = __builtin_nontemporal_load(xs + j);
    int   i0  = __builtin_nontemporal_load(perm + j);
    float lad = 0.0f;
    spline_lds<128, 8>(tab + 0,   xv, lad);  // ceil(log2(129)) = 8
    spline_lds<64, 7> (tab + 512, xv, lad);  // ceil(log2(65))  = 7
    spline_lds<32, 6> (tab + 768, xv, lad);  // ceil(log2(33))  = 6
    __builtin_nontemporal_store(xv, out + i0);
    __builtin_nontemporal_store(lad, out + n + i0);
  }
}

// ---------------- launcher ----------------
extern "C" void kernel_launch(void* const* d_in, const int* in_sizes, int n_in,
                              void* d_out, int out_size, void* d_ws, size_t ws_size,
                              hipStream_t stream) {
  const float* x      = (const float*)d_in[0];
  const int*   gix    = (const int*)d_in[1];
  const float* unnorm = (const float*)d_in[2];
  float*       out    = (float*)d_out;

  const int n = in_sizes[0];
  const int G = in_sizes[2] / PARAM_STRIDE;  // 5000

  // ws partition
  float* tabs   = (float*)d_ws;                   // G * 896 f32
  int*   counts = (int*)(tabs + (size_t)G * TAB_STRIDE);
  int*   starts = counts + G;                     // G + 1
  int*   cursor = starts + G + 1;                 // G
  int*   perm   = cursor + G;                     // n
  float* xs     = (float*)(perm + n);             // n

  const int eb = (n + 255) / 256;
  const int gb = (G + 255) / 256;

  qspline_zero_counts<<<gb, 256, 0, stream>>>(counts, G);
  qspline_hist<<<eb, 256, 0, stream>>>(gix, counts, n);
  qspline_scan_starts<<<1, 1024, 0, stream>>>(counts, starts, cursor, G);
  qspline_scatter<<<eb, 256, 0, stream>>>(x, gix, cursor, perm, xs, n);
  qspline_build_tables<<<G, 128, 0, stream>>>(unnorm, tabs, G);
  qspline_eval_sorted<<<G, 256, 0, stream>>>(xs, perm, starts, tabs, out, n);
}